// AMN_73117523247520
// MI455X (gfx1250) — compile-verified
//
#include <hip/hip_runtime.h>

typedef __attribute__((ext_vector_type(2))) float v2f;
typedef __attribute__((ext_vector_type(8))) float v8f;

#define U_DIM 128
#define T_DIM 256
#define N_DIM 1024
#define TN (T_DIM * N_DIM)          // 262144
#define NTILES (TN / 16)            // 16384 tiles of 16 contiguous (t,n) columns
#define MAIN_BLOCKS 512

// workspace layout (floats)
#define WS_W 0        // [0..127]   w[j] = 3 * gated column sums of conn
#define WS_ISUM 128   // sum(input_spikes)
#define WS_FSUM 129   // sum(final) pre-boost
#define WS_PART 130   // [130 .. 130+MAIN_BLOCKS) per-block partial sums

// ---------------------------------------------------------------------------
// Kernel 1: w[j] (threshold-gated inbound weights) + deterministic
// sum(input_spikes). Tiny: 16K + 256K reads, one block.
// ---------------------------------------------------------------------------
__global__ __launch_bounds__(256)
void amn_prep_kernel(const float* __restrict__ input_spikes,
                     const float* __restrict__ conn,
                     float* __restrict__ ws) {
    const int tid = threadIdx.x;
    if (tid < U_DIM) {
        float s = 0.0f;
        for (int i = 0; i < U_DIM; ++i) {
            float c = conn[i * U_DIM + tid];      // sum over axis 0 (inbound i)
            s += (c > 0.1f) ? c : 0.0f;
        }
        ws[WS_W + tid] = 3.0f * s;                // CONNECTION_STRENGTH fused in
    }
    // deterministic mean(input_spikes): strided accumulate + shared tree
    float s = 0.0f;
    for (int i = tid; i < TN; i += 256) s += input_spikes[i];
    __shared__ float red[256];
    red[tid] = s;
    __syncthreads();
    for (int off = 128; off > 0; off >>= 1) {
        if (tid < off) red[tid] += red[tid + off];
        __syncthreads();
    }
    if (tid == 0) ws[WS_ISUM] = red[0];
}

// ---------------------------------------------------------------------------
// Kernel 2: the HBM-bound pass. final[tn] = 0.5 * (w . uo[:,tn]) + 1.5*target.
// j-reduction expressed as 32 chained V_WMMA_F32_16X16X4_F32 per 16-wide tile:
//   A[m,k] = w[j0+k]  (rows identical; K split K=2*half / K=2*half+1 per ISA)
//   B[k,n] = uo[j0+k, base+n]  (row K striped across lanes N=lane&15)
//   D rows identical -> result for column n read from c[0].
// Also produces a deterministic per-block partial sum for the global mean.
// ---------------------------------------------------------------------------
__global__ __launch_bounds__(256)
void amn_gemv_wmma_kernel(const float* __restrict__ uo,
                          const float* __restrict__ target,
                          const float* __restrict__ ws,
                          float* __restrict__ out,
                          float* __restrict__ partials) {
    const int lane = threadIdx.x & 31;
    const int half = lane >> 4;       // 0 => lanes 0-15, 1 => lanes 16-31
    const int col  = lane & 15;       // N index within the 16-wide tile
    const int wave = blockIdx.x * (blockDim.x >> 5) + (threadIdx.x >> 5);
    const int nwaves = gridDim.x * (blockDim.x >> 5);

    // Preload A fragments for all 32 K-blocks (w is tile-invariant).
    // VGPR0 = A[., K=2*half], VGPR1 = A[., K=2*half+1].
    v2f a[32];
#pragma unroll
    for (int jb = 0; jb < 32; ++jb) {
        a[jb].x = ws[WS_W + jb * 4 + 2 * half];
        a[jb].y = ws[WS_W + jb * 4 + 2 * half + 1];
    }

    float tsum = 0.0f;
    for (int tile = wave; tile < NTILES; tile += nwaves) {   // wave-uniform loop
        const int base = tile << 4;

        // prefetch next tile's leading row (gfx1250 global_prefetch_b8)
        if (tile + nwaves < NTILES)
            __builtin_prefetch(uo + ((tile + nwaves) << 4) + col, 0, 0);

        v8f c = {};
        const float* p = uo + (size_t)(2 * half) * TN + base + col;
#pragma unroll
        for (int jb = 0; jb < 32; ++jb) {
            v2f b;
            b.x = p[0];                 // row K = 4*jb + 2*half
            b.y = p[TN];                // row K = 4*jb + 2*half + 1
            p += (size_t)4 * TN;
            c = __builtin_amdgcn_wmma_f32_16x16x4_f32(
                    /*neg_a=*/false, a[jb], /*neg_b=*/false, b,
                    /*c_mod=*/(short)0, c, /*reuse_a=*/false, /*reuse_b=*/false);
        }

        float r = c[0] * 0.5f + 1.5f * target[base + col];
        if (half == 0) {                // one unique copy per column
            out[base + col] = r;
            tsum += r;
        }
    }

    // deterministic per-block reduction of the pre-boost sum
    __shared__ float red[256];
    red[threadIdx.x] = tsum;
    __syncthreads();
    for (int off = 128; off > 0; off >>= 1) {
        if (threadIdx.x < off) red[threadIdx.x] += red[threadIdx.x + off];
        __syncthreads();
    }
    if (threadIdx.x == 0) partials[blockIdx.x] = red[0];
}

// ---------------------------------------------------------------------------
// Kernel 3: fixed-order reduction of the MAIN_BLOCKS partials -> ws[WS_FSUM].
// ---------------------------------------------------------------------------
__global__ __launch_bounds__(256)
void amn_reduce_kernel(float* __restrict__ ws) {
    const int tid = threadIdx.x;
    float s = 0.0f;
    for (int i = tid; i < MAIN_BLOCKS; i += 256) s += ws[WS_PART + i];
    __shared__ float red[256];
    red[tid] = s;
    __syncthreads();
    for (int off = 128; off > 0; off >>= 1) {
        if (tid < off) red[tid] += red[tid + off];
        __syncthreads();
    }
    if (tid == 0) ws[WS_FSUM] = red[0];
}

// ---------------------------------------------------------------------------
// Kernel 4: conditional boost. mean/boost recomputed per-thread from the two
// workspace scalars (uniform -> SALU-friendly).
// ---------------------------------------------------------------------------
__global__ __launch_bounds__(256)
void amn_boost_kernel(const float* __restrict__ rand_bias,
                      const float* __restrict__ ws,
                      float* __restrict__ out) {
    const float mean = ws[WS_FSUM] * (1.0f / TN);
    const float input_rate = ws[WS_ISUM] * (1.0f / TN) * 1000.0f;
    const float target_mean = (input_rate + 20.0f) * 0.01f;
    const float boost = fmaxf(0.0f, target_mean - mean);
    const int i = blockIdx.x * blockDim.x + threadIdx.x;
    if (i < TN && mean < 0.2f)
        out[i] += rand_bias[i] * boost * 2.0f;
}

extern "C" void kernel_launch(void* const* d_in, const int* in_sizes, int n_in,
                              void* d_out, int out_size, void* d_ws, size_t ws_size,
                              hipStream_t stream) {
    const float* input_spikes = (const float*)d_in[0];   // [T,N]
    const float* unit_outputs = (const float*)d_in[1];   // [U,T,N]
    const float* conn         = (const float*)d_in[2];   // [U,U]
    const float* target       = (const float*)d_in[3];   // [T,N]
    const float* rand_bias    = (const float*)d_in[4];   // [T,N]
    float* out = (float*)d_out;
    float* ws  = (float*)d_ws;

    amn_prep_kernel<<<1, 256, 0, stream>>>(input_spikes, conn, ws);
    amn_gemv_wmma_kernel<<<MAIN_BLOCKS, 256, 0, stream>>>(
        unit_outputs, target, ws, out, ws + WS_PART);
    amn_reduce_kernel<<<1, 256, 0, stream>>>(ws);
    amn_boost_kernel<<<(TN + 255) / 256, 256, 0, stream>>>(rand_bias, ws, out);
}